// _Query_Guided_Attention_72370198938044
// MI455X (gfx1250) — compile-verified
//
#include <hip/hip_runtime.h>

typedef __attribute__((ext_vector_type(2))) float v2f;
typedef __attribute__((ext_vector_type(8))) float v8f;

#define IN_CH  256
#define OCH    128
#define BATCH  32
#define WW     64
#define HW     4096      // 64*64
#define NPOOL  1024      // 32*32

// ---------------------------------------------------------------------------
// gfx1250 async copy global->LDS (ASYNCcnt-tracked), per CDNA5 ISA.
// Each lane moves 16 bytes; LDS dest offset in a VGPR, 64-bit global addr.
// ---------------------------------------------------------------------------
__device__ __forceinline__ void async_ld16(unsigned int lds_off,
                                           unsigned long long gaddr) {
  asm volatile("global_load_async_to_lds_b128 %0, %1, off"
               :: "v"(lds_off), "v"(gaddr) : "memory");
}
__device__ __forceinline__ void wait_async0() {
  asm volatile("s_wait_asynccnt 0x0" ::: "memory");
}

// WMMA fragment layouts (wave32, V_WMMA_F32_16X16X4_F32):
//   A (16x4, MxK): lane l holds {A[m][k0], A[m][k0+1]}, m=l&15, k0=2*(l>>4)
//   B (4x16, KxN): lane l holds {B[k0][n], B[k0+1][n]},  n=l&15, k0=2*(l>>4)
//   C/D (16x16):   VGPR r -> M=r (lanes 0-15), M=r+8 (lanes 16-31), N=lane&15

// ---------------------------------------------------------------------------
// Projection: out[b][o][p] = sum_c W[o][c]*x[b][c][p] + bias[o]
// One block (8 waves) = one (b, 64-pixel strip); wave w handles o-tile w.
// x strip staged once per block through LDS, double-buffered async chunks
// of 16(k) x 64(p). Padded stride 80 floats -> B-frag reads conflict-free.
// ---------------------------------------------------------------------------
#define PSTR 80

__global__ __launch_bounds__(256)
void proj_kernel(const float* __restrict__ x, const float* __restrict__ w,
                 const float* __restrict__ bias, float* __restrict__ out) {
  __shared__ float sx[2][16 * PSTR];

  const int tid  = (int)threadIdx.x;
  const int warp = tid >> 5;
  const int lane = tid & 31;
  const int m    = lane & 15;
  const int half = lane >> 4;

  const int b  = (int)blockIdx.x >> 6;   // 64 blocks per batch
  const int pb = (int)blockIdx.x & 63;
  const int p0 = pb * 64;
  const int o0 = warp * 16;              // 8 waves cover all 128 o

  const float* xb = x + (size_t)b * IN_CH * HW + p0;

  // Cooperative chunk copy: thread t moves x[k0+kk][p0+pp .. pp+3]
  const int kk = tid >> 4;               // 0..15 (k row in chunk)
  const int pp = (tid & 15) * 4;         // 0..60 (pixel col, float4)
  const unsigned long long gbase =
      (unsigned long long)(size_t)(xb + (size_t)kk * HW + pp);
  const unsigned int lbase =
      (unsigned int)(size_t)(&sx[0][kk * PSTR + pp]);
  const unsigned int lsz = (unsigned int)(16 * PSTR * 4);

  async_ld16(lbase, gbase);              // chunk 0 -> buffer 0

  v8f acc0 = {}, acc1 = {}, acc2 = {}, acc3 = {};

  for (int ch = 0; ch < 16; ++ch) {      // 16 K-chunks of 16
    wait_async0();                       // chunk ch landed (this wave's copies)
    __syncthreads();                     // all waves: copies done + prev compute done
    if (ch + 1 < 16)
      async_ld16(lbase + (((ch + 1) & 1) ? lsz : 0u),
                 gbase + (unsigned long long)(ch + 1) * 16ull * HW * 4ull);

    const float* sp = &sx[ch & 1][0];
    const int k0 = ch * 16;
#pragma unroll
    for (int kg = 0; kg < 4; ++kg) {
      const int ka = k0 + kg * 4 + 2 * half;
      v2f a = *(const v2f*)(w + (size_t)(o0 + m) * IN_CH + ka);  // A: weights
      const float* r0 = sp + (kg * 4 + 2 * half) * PSTR + m;     // B: x via LDS
      const float* r1 = r0 + PSTR;
      v2f b0, b1, b2, b3;
      b0.x = r0[0];  b0.y = r1[0];
      b1.x = r0[16]; b1.y = r1[16];
      b2.x = r0[32]; b2.y = r1[32];
      b3.x = r0[48]; b3.y = r1[48];
      acc0 = __builtin_amdgcn_wmma_f32_16x16x4_f32(false, a, false, b0, (short)0, acc0, false, false);
      acc1 = __builtin_amdgcn_wmma_f32_16x16x4_f32(false, a, false, b1, (short)0, acc1, false, false);
      acc2 = __builtin_amdgcn_wmma_f32_16x16x4_f32(false, a, false, b2, (short)0, acc2, false, false);
      acc3 = __builtin_amdgcn_wmma_f32_16x16x4_f32(false, a, false, b3, (short)0, acc3, false, false);
    }
  }

  float* ob = out + (size_t)b * OCH * HW;
#pragma unroll
  for (int r = 0; r < 8; ++r) {
    const int o = o0 + 8 * half + r;
    const float bv = bias[o];
    float* orow = ob + (size_t)o * HW + p0 + m;
    orow[0]  = acc0[r] + bv;
    orow[16] = acc1[r] + bv;
    orow[32] = acc2[r] + bv;
    orow[48] = acc3[r] + bv;
  }
}

// ---------------------------------------------------------------------------
// 2x2 max pool, emitting WMMA-fragment-ready pair-interleaved layout:
//   phipF[b][kp][n][2],  kp = o>>1, j = o&1  ->  {phi[2kp][n], phi[2kp+1][n]}
//   are 8 contiguous bytes (one ds_load_b64 per B-fragment downstream).
// Thread mapping: j fastest -> fully coalesced writes.
// ---------------------------------------------------------------------------
__global__ __launch_bounds__(256)
void pool_kernel(const float* __restrict__ phix, float* __restrict__ phipF) {
  const int idx = (int)(blockIdx.x * blockDim.x + threadIdx.x); // B*OCH*NPOOL
  const int j   = idx & 1;               // low/high channel of the pair
  const int n   = (idx >> 1) & (NPOOL - 1);
  const int bkp = idx >> 11;             // b*64 + kp
  const int o   = ((bkp & 63) << 1) | j;
  const int b   = bkp >> 6;
  const int hp = n >> 5, wp = n & 31;
  const float* src = phix + ((size_t)b * OCH + o) * HW + (size_t)(hp * 2) * WW + wp * 2;
  const float m0 = fmaxf(src[0], src[1]);
  const float m1 = fmaxf(src[WW], src[WW + 1]);
  phipF[((size_t)bkp * NPOOL + n) * 2 + j] = fmaxf(m0, m1);
}

// ---------------------------------------------------------------------------
// Affinity + row max:
//   attn[b][p] = (1/NPOOL) * max_n sum_o theta[b][o][p] * phipF[b][o>>1][n][o&1]
// One block (8 waves) = one (b, 128-row p-group); wave w -> p-tile w.
// phi n-chunks (64 kp-rows x 32 floats = 8 KB) staged async into LDS,
// shared by all 8 waves, double-buffered. B-frag = one aligned ds_load_b64
// (half=0 lanes hit banks 0-31, half=1 lanes banks 32-63: conflict-free).
// Theta A-frags hoisted in registers, reused across all 64 n-tiles.
// ---------------------------------------------------------------------------
__global__ __launch_bounds__(256)
void affinity_kernel(const float* __restrict__ theta, const float* __restrict__ phipF,
                     float* __restrict__ attn) {
  __shared__ float sphi[2][64 * 32];     // 2 x 8 KB

  const int tid  = (int)threadIdx.x;
  const int warp = tid >> 5;
  const int lane = tid & 31;
  const int m    = lane & 15;
  const int half = lane >> 4;

  const int b   = (int)blockIdx.x >> 5;  // 32 blocks per batch
  const int ptg = (int)blockIdx.x & 31;
  const int p0  = (ptg * 8 + warp) * 16;

  const float* tb  = theta + (size_t)b * OCH * HW;
  const float* phb = phipF + (size_t)b * 64 * NPOOL * 2;

  // A[m=p][k=o] = theta[o][p0+m]; 32 frags cover K=128, reused for all n.
  v2f afr[32];
#pragma unroll
  for (int j = 0; j < 32; ++j) {
    const float* r = tb + (size_t)(4 * j + 2 * half) * HW + p0 + m;
    afr[j].x = r[0];
    afr[j].y = r[HW];
  }

  // Cooperative chunk copy: 64 kp-rows x 32 floats = 512 float4; thread t
  // moves float4 #t and #(t+256).  q -> row kp=q>>3, float4 col q&7.
  const int q0 = tid, q1 = tid + 256;
  const unsigned long long gb0 =
      (unsigned long long)(size_t)(phb + (size_t)(q0 >> 3) * (NPOOL * 2) + (q0 & 7) * 4);
  const unsigned long long gb1 =
      (unsigned long long)(size_t)(phb + (size_t)(q1 >> 3) * (NPOOL * 2) + (q1 & 7) * 4);
  const unsigned int lb0 = (unsigned int)(size_t)(&sphi[0][(q0 >> 3) * 32 + (q0 & 7) * 4]);
  const unsigned int lb1 = (unsigned int)(size_t)(&sphi[0][(q1 >> 3) * 32 + (q1 & 7) * 4]);
  const unsigned int lsz = (unsigned int)(64 * 32 * 4);

  async_ld16(lb0, gb0);                  // chunk 0 -> buffer 0
  async_ld16(lb1, gb1);

  v8f runmax;
#pragma unroll
  for (int r = 0; r < 8; ++r) runmax[r] = -3.402823466e38f;

  for (int nt = 0; nt < 64; ++nt) {
    wait_async0();
    __syncthreads();
    if (nt + 1 < 64) {
      const unsigned long long go = (unsigned long long)(nt + 1) * 128ull; // 32 floats
      const unsigned int lo = ((nt + 1) & 1) ? lsz : 0u;
      async_ld16(lb0 + lo, gb0 + go);
      async_ld16(lb1 + lo, gb1 + go);
    }

    const float* sp = &sphi[nt & 1][0];
    v8f acc = {};
#pragma unroll
    for (int j = 0; j < 32; ++j) {
      // pair {phi[4j+2h][n0+m], phi[4j+2h+1][n0+m]} is one aligned b64
      v2f bf = *(const v2f*)(sp + (2 * j + half) * 32 + 2 * m);
      acc = __builtin_amdgcn_wmma_f32_16x16x4_f32(false, afr[j], false, bf,
                                                  (short)0, acc, false, false);
    }
#pragma unroll
    for (int r = 0; r < 8; ++r) runmax[r] = fmaxf(runmax[r], acc[r]);
  }

  // Cross-lane max within each 16-lane half (masks < 16 never cross halves).
#pragma unroll
  for (int mask = 1; mask <= 8; mask <<= 1) {
#pragma unroll
    for (int r = 0; r < 8; ++r)
      runmax[r] = fmaxf(runmax[r], __shfl_xor(runmax[r], mask, 32));
  }

  // lane 0 -> rows p0..p0+7 (component r), lane 16 -> p0+8..p0+15
  if (m == 0) {
    float* dst = attn + (size_t)b * HW + p0 + 8 * half;
#pragma unroll
    for (int r = 0; r < 8; ++r) dst[r] = runmax[r] * (1.0f / (float)NPOOL);
  }
}

// ---------------------------------------------------------------------------
// out[b][c][hw] = x[b][c][hw] * attn[b][hw]   (float4 vectorized)
// ---------------------------------------------------------------------------
__global__ __launch_bounds__(256)
void mul_kernel(const float* __restrict__ x, const float* __restrict__ attn,
                float* __restrict__ out) {
  const size_t idx = (size_t)blockIdx.x * blockDim.x + threadIdx.x;
  const size_t e0  = idx * 4;
  const int hw = (int)(e0 & (HW - 1));
  const int b  = (int)(e0 >> 20);        // / (IN_CH*HW) = / 2^20
  const float4 xv = *(const float4*)(x + e0);
  const float4 av = *(const float4*)(attn + (size_t)b * HW + hw);
  float4 o;
  o.x = xv.x * av.x;
  o.y = xv.y * av.y;
  o.z = xv.z * av.z;
  o.w = xv.w * av.w;
  *(float4*)(out + e0) = o;
}

extern "C" void kernel_launch(void* const* d_in, const int* in_sizes, int n_in,
                              void* d_out, int out_size, void* d_ws, size_t ws_size,
                              hipStream_t stream) {
  const float* x     = (const float*)d_in[0];
  const float* xg    = (const float*)d_in[1];
  const float* tw    = (const float*)d_in[2];
  const float* tbias = (const float*)d_in[3];
  const float* pw    = (const float*)d_in[4];
  const float* pbias = (const float*)d_in[5];
  float* out = (float*)d_out;

  float* ws    = (float*)d_ws;
  float* theta = ws;                                    // 32*128*4096 floats
  float* phix  = theta + (size_t)BATCH * OCH * HW;      // 32*128*4096 floats
  float* phipF = phix  + (size_t)BATCH * OCH * HW;      // 32*64*1024*2 floats
  float* attn  = phipF + (size_t)BATCH * OCH * NPOOL;   // 32*4096 floats

  // one block per (b, 64-pixel strip): 32*64 = 2048 blocks
  proj_kernel<<<2048, 256, 0, stream>>>(x,  tw, tbias, theta);
  proj_kernel<<<2048, 256, 0, stream>>>(xg, pw, pbias, phix);
  // 32*128*1024 pooled outputs (pair-interleaved layout)
  pool_kernel<<<(BATCH * OCH * NPOOL) / 256, 256, 0, stream>>>(phix, phipF);
  // one block per (b, 128-row p-group): 32*32 = 1024 blocks
  affinity_kernel<<<1024, 256, 0, stream>>>(theta, phipF, attn);
  // 33.5M floats / 4 per thread / 256 per block
  mul_kernel<<<(BATCH * IN_CH * HW / 4) / 256, 256, 0, stream>>>(x, attn, out);
}